// GATNet_85555748537152
// MI455X (gfx1250) — compile-verified
//
#include <hip/hip_runtime.h>
#include <math.h>

typedef __attribute__((ext_vector_type(2))) float v2f;
typedef __attribute__((ext_vector_type(8))) float v8f;

#define NEG_SLOPE 0.2f

// ---------------------------------------------------------------------------
// GEMM: C[M x NC] = A[M x K] @ B[K x NC], all f32, via V_WMMA_F32_16X16X4_F32.
// One wave32 computes one 16x16 tile of C; K-loop in steps of 4.
// A-layout (16x4 f32): lanes 0-15 hold {K=k0,k0+1}, lanes 16-31 hold {K=k0+2,k0+3}
// B-layout (4x16 f32): symmetric (row striped across lanes within a VGPR)
// C-layout (16x16 f32): VGPR j -> row half*8+j, col = lane&15
// ---------------------------------------------------------------------------
__global__ __launch_bounds__(128) void gemm_f32_wmma_kernel(
    const float* __restrict__ A, const float* __restrict__ B, float* __restrict__ C,
    int M, int K, int NC, int tiles_n, int total_tiles) {
  int wave = threadIdx.x >> 5;
  int lane = threadIdx.x & 31;
  int tile = blockIdx.x * 4 + wave;
  if (tile >= total_tiles) return;            // wave-uniform branch (EXEC stays all-1s)
  int tm = tile / tiles_n;
  int tn = tile % tiles_n;
  int half = lane >> 4;                       // 0: K=k0..k0+1, 1: K=k0+2..k0+3
  int l15  = lane & 15;

  const float* aptr = A + (size_t)(tm * 16 + l15) * K + half * 2;
  const float* bptr = B + (size_t)(half * 2) * NC + tn * 16 + l15;

  v8f acc = {};
  for (int k0 = 0; k0 < K; k0 += 4) {
    v2f av = { aptr[0], aptr[1] };            // contiguous -> global_load_b64
    v2f bv = { bptr[0], bptr[NC] };
    acc = __builtin_amdgcn_wmma_f32_16x16x4_f32(
        /*neg_a=*/false, av, /*neg_b=*/false, bv,
        /*c_mod=*/(short)0, acc, /*reuse_a=*/false, /*reuse_b=*/false);
    aptr += 4;
    bptr += (size_t)4 * NC;
  }

  float* cptr = C + (size_t)(tm * 16 + half * 8) * NC + tn * 16 + l15;
#pragma unroll
  for (int j = 0; j < 8; ++j) cptr[(size_t)j * NC] = acc[j];
}

// alpha_s[n,h] = dot(h[n,h,:], a_src[h,:]);  alpha_d likewise. One thread per (n,h).
__global__ __launch_bounds__(256) void alpha_kernel(
    const float* __restrict__ h, const float* __restrict__ a_src,
    const float* __restrict__ a_dst, float* __restrict__ as, float* __restrict__ ad,
    int n, int heads) {
  int idx = blockIdx.x * blockDim.x + threadIdx.x;
  if (idx >= n * heads) return;
  int head = idx % heads;
  const float4* hp = (const float4*)(h + (size_t)idx * 64);
  const float4* s  = (const float4*)(a_src + head * 64);
  const float4* d  = (const float4*)(a_dst + head * 64);
  float accs = 0.f, accd = 0.f;
#pragma unroll
  for (int k = 0; k < 16; ++k) {
    float4 v = hp[k], sv = s[k], dv = d[k];
    accs += v.x * sv.x + v.y * sv.y + v.z * sv.z + v.w * sv.w;
    accd += v.x * dv.x + v.y * dv.y + v.z * dv.z + v.w * dv.w;
  }
  as[idx] = accs;
  ad[idx] = accd;
}

__global__ __launch_bounds__(256) void fill_kernel(float* __restrict__ p,
                                                   long long count, float val) {
  long long i = (long long)blockIdx.x * blockDim.x + threadIdx.x;
  if (i < count) p[i] = val;
}

// float atomic-max via signed/unsigned integer atomics (always available)
__device__ __forceinline__ void atomic_max_float(float* addr, float val) {
  if (val >= 0.f) atomicMax((int*)addr, __float_as_int(val));
  else            atomicMin((unsigned int*)addr, __float_as_uint(val));
}

__device__ __forceinline__ void get_edge(const int* __restrict__ ei, int E, int i,
                                         int& src, int& dst) {
  if (i < E) { src = ei[i]; dst = ei[E + i]; }   // edge_index rows [0]=src, [1]=dst
  else       { src = i - E; dst = i - E; }       // self loops appended
}

__global__ __launch_bounds__(256) void edge_max_kernel(
    const int* __restrict__ ei, int E, int n,
    const float* __restrict__ as, const float* __restrict__ ad,
    float* __restrict__ m, int heads) {
  int i = blockIdx.x * blockDim.x + threadIdx.x;
  if (i >= E + n) return;
  int src, dst; get_edge(ei, E, i, src, dst);
  for (int h = 0; h < heads; ++h) {
    float e = as[src * heads + h] + ad[dst * heads + h];
    e = (e > 0.f) ? e : NEG_SLOPE * e;
    atomic_max_float(&m[dst * heads + h], e);
  }
}

__global__ __launch_bounds__(256) void edge_sum_kernel(
    const int* __restrict__ ei, int E, int n,
    const float* __restrict__ as, const float* __restrict__ ad,
    const float* __restrict__ m, float* __restrict__ den, int heads) {
  int i = blockIdx.x * blockDim.x + threadIdx.x;
  if (i >= E + n) return;
  int src, dst; get_edge(ei, E, i, src, dst);
  for (int h = 0; h < heads; ++h) {
    float e = as[src * heads + h] + ad[dst * heads + h];
    e = (e > 0.f) ? e : NEG_SLOPE * e;
    atomicAdd(&den[dst * heads + h], __expf(e - m[dst * heads + h]));
  }
}

// One wave32 per (edge, head): lanes cover dims {lane, lane+32}.
// h[src] gather and agg[dst] atomic scatter are fully coalesced (2x128B per wave).
// Per-edge scalars computed from wave-uniform addresses -> scalarized by compiler.
__global__ __launch_bounds__(256) void edge_agg_kernel(
    const int* __restrict__ ei, int E, int n,
    const float* __restrict__ as, const float* __restrict__ ad,
    const float* __restrict__ m, const float* __restrict__ den,
    const float* __restrict__ h, float* __restrict__ agg, int heads) {
  long long wid = ((long long)blockIdx.x * blockDim.x + threadIdx.x) >> 5;
  int lane = threadIdx.x & 31;
  long long tot = (long long)(E + n) * heads;
  if (wid >= tot) return;
  int hd = (int)(wid % heads);
  int i  = (int)(wid / heads);
  int src, dst; get_edge(ei, E, i, src, dst);
  float e = as[src * heads + hd] + ad[dst * heads + hd];
  e = (e > 0.f) ? e : NEG_SLOPE * e;
  float alpha = __expf(e - m[dst * heads + hd]) / den[dst * heads + hd];
  const float* hs = h + ((size_t)src * heads + hd) * 64;
  float* op = agg + ((size_t)dst * heads + hd) * 64;
  atomicAdd(op + lane,      hs[lane]      * alpha);
  atomicAdd(op + lane + 32, hs[lane + 32] * alpha);
}

// in-place: a = elu(a + bias[c])
__global__ __launch_bounds__(256) void elu_bias_kernel(
    float* __restrict__ a, const float* __restrict__ bias, long long total, int C) {
  long long i = (long long)blockIdx.x * blockDim.x + threadIdx.x;
  if (i >= total) return;
  float v = a[i] + bias[(int)(i % C)];
  a[i] = (v > 0.f) ? v : (__expf(v) - 1.f);
}

__global__ __launch_bounds__(256) void final_kernel(
    const float* __restrict__ hfin, const float* __restrict__ lw,
    const float* __restrict__ lb, float* __restrict__ out, int n) {
  int i = blockIdx.x * blockDim.x + threadIdx.x;
  if (i >= n) return;
  const float4* hp = (const float4*)(hfin + (size_t)i * 64);
  float o0 = 0.f, o1 = 0.f;
#pragma unroll
  for (int d = 0; d < 16; ++d) {
    float4 v = hp[d];
    o0 += v.x * lw[(d*4+0)*2] + v.y * lw[(d*4+1)*2] + v.z * lw[(d*4+2)*2] + v.w * lw[(d*4+3)*2];
    o1 += v.x * lw[(d*4+0)*2+1] + v.y * lw[(d*4+1)*2+1] + v.z * lw[(d*4+2)*2+1] + v.w * lw[(d*4+3)*2+1];
  }
  out[i * 2 + 0] = o0 + lb[0];
  out[i * 2 + 1] = o1 + lb[1];
}

static inline int cdiv(long long a, long long b) { return (int)((a + b - 1) / b); }

extern "C" void kernel_launch(void* const* d_in, const int* in_sizes, int n_in,
                              void* d_out, int out_size, void* d_ws, size_t ws_size,
                              hipStream_t stream) {
  const float* x      = (const float*)d_in[0];
  const int*   ei     = (const int*)  d_in[1];
  const float* W1     = (const float*)d_in[2];
  const float* a_src1 = (const float*)d_in[3];
  const float* a_dst1 = (const float*)d_in[4];
  const float* b1     = (const float*)d_in[5];
  const float* W2     = (const float*)d_in[6];
  const float* a_src2 = (const float*)d_in[7];
  const float* a_dst2 = (const float*)d_in[8];
  const float* b2     = (const float*)d_in[9];
  const float* lin_w  = (const float*)d_in[10];
  const float* lin_b  = (const float*)d_in[11];
  float* out = (float*)d_out;

  int n = in_sizes[0] / 128;   // 100000 (divisible by 16)
  int E = in_sizes[1] / 2;     // 3200000
  int tot = E + n;             // edges + self loops
  (void)n_in; (void)out_size; (void)ws_size;

  // workspace layout (floats)
  float* ws = (float*)d_ws;
  size_t o = 0;
  float* h1   = ws + o; o += (size_t)n * 256;   // layer-1 features
  float* agg1 = ws + o; o += (size_t)n * 256;   // layer-1 aggregated output
  float* as1  = ws + o; o += (size_t)n * 4;
  float* ad1  = ws + o; o += (size_t)n * 4;
  float* m1   = ws + o; o += (size_t)n * 4;
  float* den1 = ws + o; o += (size_t)n * 4;
  float* as2  = ws + o; o += (size_t)n;
  float* ad2  = ws + o; o += (size_t)n;
  float* m2   = ws + o; o += (size_t)n;
  float* den2 = ws + o; o += (size_t)n;
  float* h2   = h1;                             // reuse: h1 dead after layer-1 agg
  float* agg2 = h1 + (size_t)n * 64;            // disjoint sub-region of h1 block

  // ---- Layer 1: h1 = x @ W1  (M=n, K=128, NC=256) ----
  {
    int tiles_n = 256 / 16, total_tiles = (n / 16) * tiles_n;
    gemm_f32_wmma_kernel<<<cdiv(total_tiles, 4), 128, 0, stream>>>(
        x, W1, h1, n, 128, 256, tiles_n, total_tiles);
  }
  alpha_kernel<<<cdiv((long long)n * 4, 256), 256, 0, stream>>>(
      h1, a_src1, a_dst1, as1, ad1, n, 4);
  fill_kernel<<<cdiv((long long)n * 4, 256), 256, 0, stream>>>(m1, (long long)n * 4, -1e30f);
  fill_kernel<<<cdiv((long long)n * 4, 256), 256, 0, stream>>>(den1, (long long)n * 4, 0.f);
  fill_kernel<<<cdiv((long long)n * 256, 256), 256, 0, stream>>>(agg1, (long long)n * 256, 0.f);
  edge_max_kernel<<<cdiv(tot, 256), 256, 0, stream>>>(ei, E, n, as1, ad1, m1, 4);
  edge_sum_kernel<<<cdiv(tot, 256), 256, 0, stream>>>(ei, E, n, as1, ad1, m1, den1, 4);
  edge_agg_kernel<<<cdiv((long long)tot * 4 * 32, 256), 256, 0, stream>>>(
      ei, E, n, as1, ad1, m1, den1, h1, agg1, 4);
  elu_bias_kernel<<<cdiv((long long)n * 256, 256), 256, 0, stream>>>(
      agg1, b1, (long long)n * 256, 256);

  // ---- Layer 2: h2 = elu(agg1) @ W2  (M=n, K=256, NC=64) ----
  {
    int tiles_n = 64 / 16, total_tiles = (n / 16) * tiles_n;
    gemm_f32_wmma_kernel<<<cdiv(total_tiles, 4), 128, 0, stream>>>(
        agg1, W2, h2, n, 256, 64, tiles_n, total_tiles);
  }
  alpha_kernel<<<cdiv(n, 256), 256, 0, stream>>>(h2, a_src2, a_dst2, as2, ad2, n, 1);
  fill_kernel<<<cdiv(n, 256), 256, 0, stream>>>(m2, n, -1e30f);
  fill_kernel<<<cdiv(n, 256), 256, 0, stream>>>(den2, n, 0.f);
  fill_kernel<<<cdiv((long long)n * 64, 256), 256, 0, stream>>>(agg2, (long long)n * 64, 0.f);
  edge_max_kernel<<<cdiv(tot, 256), 256, 0, stream>>>(ei, E, n, as2, ad2, m2, 1);
  edge_sum_kernel<<<cdiv(tot, 256), 256, 0, stream>>>(ei, E, n, as2, ad2, m2, den2, 1);
  edge_agg_kernel<<<cdiv((long long)tot * 32, 256), 256, 0, stream>>>(
      ei, E, n, as2, ad2, m2, den2, h2, agg2, 1);
  elu_bias_kernel<<<cdiv((long long)n * 64, 256), 256, 0, stream>>>(
      agg2, b2, (long long)n * 64, 64);

  // ---- Final linear 64 -> 2 ----
  final_kernel<<<cdiv(n, 256), 256, 0, stream>>>(agg2, lin_w, lin_b, out, n);
}